// SlidingMean_63445256896738
// MI455X (gfx1250) — compile-verified
//
#include <hip/hip_runtime.h>
#include <cstdint>
#include <cstddef>

// ---------------------------------------------------------------------------
// (x - boxmean_79x69(x)) / 5 over 32x512x512x1 fp32, SAME zero padding.
//
// Memory-bound (~67 MB effective HBM traffic => ~3us floor at 23.3 TB/s).
// Fused tiled separable sliding-window sums; tiles staged into LDS with
// CDNA5 async global->LDS DMA (GLOBAL_LOAD_ASYNC_TO_LDS_B128, ASYNCcnt).
// ---------------------------------------------------------------------------

#define N_IMG 32
#define H     512
#define W     512
#define KH    79
#define KW    69
#define PT    39               // top pad  = (KH-1)/2
#define PL    34               // left pad = (KW-1)/2
#define OTH   64               // output tile height
#define OTW   64               // output tile width
#define IN_H  (OTH + KH - 1)   // 142 staged rows per tile
// Staged columns: global cols [col0-36, col0+100) -> 136 cols, 4-aligned on
// both ends so every float4 group is fully in- or out-of-bounds.
#define GOFF  36               // staging left offset (PL rounded up to mult 4)
#define SIN_W 136              // s_in row stride (floats); 136*4 B, 16B-aligned rows
#define NW4   (SIN_W / 4)      // 34 float4 groups per staged row
#define SCOL_W 137             // s_col row stride: 137 mod 64 = 9 -> conflict-free
#define TPB   256

__global__ __launch_bounds__(TPB)
void SlidingMean_63445256896738_kernel(const float* __restrict__ x,
                                       float* __restrict__ out) {
    extern __shared__ __align__(16) float smem[];
    float* s_in  = smem;                       // IN_H * SIN_W floats (halo tile)
    float* s_col = smem + IN_H * SIN_W;        // OTH  * SCOL_W floats (vertical sums)

    const int tid  = threadIdx.x;
    const int col0 = blockIdx.x * OTW;
    const int row0 = blockIdx.y * OTH;
    const int n    = blockIdx.z;

    const float* img = x + (size_t)n * H * W;

    // ---- Phase 1: stage halo tile into LDS, float4 granularity ----------
    // In-bounds groups: one async DMA b128 per lane (512 B / wave32).
    // Out-of-bounds groups (zero padding): ds_store_b128 of zeros.
    // Disjoint LDS addresses -> only counter drains + barrier needed.
    const int ngroups = IN_H * NW4;            // 4828
    for (int e = tid; e < ngroups; e += TPB) {
        const int r   = e / NW4;
        const int g   = e - r * NW4;
        const int gr  = row0 - PT + r;
        const int gc0 = col0 - GOFF + g * 4;   // 4-aligned; fully in iff < 512
        if ((unsigned)gr < (unsigned)H && (unsigned)gc0 < (unsigned)W) {
            // LDS byte offset = low 32 bits of the flat pointer (flat LDS
            // aperture keeps the allocation offset in addr[31:0]). 16B-aligned.
            unsigned lds_off = (unsigned)(uintptr_t)&s_in[e * 4];
            uint64_t gaddr   = (uint64_t)(img + (size_t)gr * W + gc0);
            asm volatile("global_load_async_to_lds_b128 %0, %1, off"
                         :: "v"(lds_off), "v"(gaddr)
                         : "memory");
        } else {
            *(float4*)&s_in[e * 4] = make_float4(0.f, 0.f, 0.f, 0.f);
        }
    }
    // Drain this wave's async DMA; barrier publishes all waves' LDS writes.
    asm volatile("s_wait_asynccnt 0x0" ::: "memory");
    __syncthreads();

    // ---- Phase 2: vertical 79-row running sums --------------------------
    // One lane per staged column; for fixed r, lanes read consecutive LDS
    // addresses -> conflict-free.
    if (tid < SIN_W) {
        const int c = tid;
        float sum = 0.0f;
        for (int r = 0; r < KH - 1; ++r)
            sum += s_in[r * SIN_W + c];
        for (int ro = 0; ro < OTH; ++ro) {
            sum += s_in[(ro + KH - 1) * SIN_W + c];  // window rows ro..ro+78
            s_col[ro * SCOL_W + c] = sum;
            sum -= s_in[ro * SIN_W + c];
        }
    }
    __syncthreads();

    // ---- Phase 3: horizontal 69-col running sums + normalize ------------
    // 4 lanes per output row, 16 outputs each, float4 stores.
    // s_col bank index = (9*r + 16*seg + o) mod 64: collision needs
    // 9*dr + 16*ds == 0 (mod 64) with |dr|<=7, |ds|<=3 -> only dr=ds=0.
    const float INV_AREA = 1.0f / (float)(KH * KW);
    const float INV_STD  = 1.0f / 5.0f;
    const int   SEG_W    = OTW / 4;            // 16

    const int r   = tid >> 2;                  // 0..63
    const int seg = tid & 3;                   // 0..3
    const int cc0 = seg * SEG_W;               // first output col of segment

    // Output col cc maps to staged col cc + (GOFF - PL) + window:
    //   window = staged cols (cc+2) .. (cc+70);  x at staged col cc + GOFF.
    const float* crow = s_col + r * SCOL_W + cc0 + (GOFF - PL);     // +2
    const float* xrow = s_in + (r + PT) * SIN_W + cc0 + GOFF;
    float4*      ovec = (float4*)(out + ((size_t)n * H + (row0 + r)) * W
                                  + col0 + cc0);                     // 64B-aligned

    float run = 0.0f;
    for (int j = 0; j < KW - 1; ++j)
        run += crow[j];

    float vals[4];
    for (int k = 0; k < SEG_W; ++k) {
        run += crow[k + KW - 1];               // window cols k+2 .. k+70
        const float mean = run * INV_AREA;
        vals[k & 3] = (xrow[k] - mean) * INV_STD;
        run -= crow[k];
        if ((k & 3) == 3)
            ovec[k >> 2] = make_float4(vals[0], vals[1], vals[2], vals[3]);
    }
}

extern "C" void kernel_launch(void* const* d_in, const int* in_sizes, int n_in,
                              void* d_out, int out_size, void* d_ws, size_t ws_size,
                              hipStream_t stream) {
    (void)in_sizes; (void)n_in; (void)out_size; (void)d_ws; (void)ws_size;

    const float* x   = (const float*)d_in[0];
    float*       out = (float*)d_out;

    const size_t shmem = (size_t)(IN_H * SIN_W + OTH * SCOL_W) * sizeof(float); // 112320 B

    // Opt in to >64KB dynamic LDS (320KB/WGP; 2 workgroups co-resident).
    (void)hipFuncSetAttribute((const void*)SlidingMean_63445256896738_kernel,
                              hipFuncAttributeMaxDynamicSharedMemorySize,
                              (int)shmem);

    dim3 grid(W / OTW, H / OTH, N_IMG);        // 8 x 8 x 32 tiles
    dim3 block(TPB);
    SlidingMean_63445256896738_kernel<<<grid, block, shmem, stream>>>(x, out);
}